// NeuralGraphOutput_13434657702340
// MI455X (gfx1250) — compile-verified
//
#include <hip/hip_runtime.h>
#include <hip/hip_bf16.h>

// Problem constants (match reference)
#define SS 4096
#define AA 128
#define DN 5
#define FAF 62
#define FBF 6
#define FPF 128
#define KP  72   // 62 atom feats + 6 bond feats + 1 count (bias row) + 3 pad -> multiple of 4

typedef float v2f __attribute__((ext_vector_type(2)));
typedef float v8f __attribute__((ext_vector_type(8)));

// ---------------------------------------------------------------------------
// Stage 0: build extended weight matrix W_ext (72 x 128):
//   rows 0..67  = W (68 x 128)
//   row  68     = bias b (multiplied by per-sample masked-atom count in GEMM)
//   rows 69..71 = 0 (K padding)
// ---------------------------------------------------------------------------
__global__ void __launch_bounds__(256) build_wext_kernel(
    const float* __restrict__ W, const float* __restrict__ b,
    float* __restrict__ wext) {
  int i = blockIdx.x * blockDim.x + threadIdx.x;   // 0 .. 72*128-1
  if (i >= KP * FPF) return;
  int r = i >> 7;          // row (K)
  int c = i & 127;         // col (FP)
  float v;
  if (r < FAF + FBF)      v = W[r * FPF + c];
  else if (r == FAF + FBF) v = b[c];
  else                     v = 0.0f;
  wext[i] = v;
}

// ---------------------------------------------------------------------------
// Stage 1: per-sample masked reduction over atoms (bandwidth-bound, ~216 MB).
// One 256-thread block per sample. Deterministic tree reductions (no float
// atomics) so timed replays are bit-identical.
// Output: msum (4096 x 72) row-major:
//   [0..61]  = sum_a mask*atoms,  [62..67] = sum_a mask * sum_d bonds,
//   [68]     = sum_a mask,        [69..71] = 0
// ---------------------------------------------------------------------------
__global__ void __launch_bounds__(256) reduce_kernel(
    const float* __restrict__ atoms, const float* __restrict__ bonds,
    const long long* __restrict__ edges, float* __restrict__ msum) {
  __shared__ float  maskL[AA];
  __shared__ float2 aPart[248];   // atoms partials: 8 row-groups x 31 float2
  __shared__ float2 bPart[240];   // bonds partials: 16 row-groups x 15 float2
  __shared__ float2 jsum[15];     // bonds per-j sums
  __shared__ float  accum[KP];

  const int s   = blockIdx.x;
  const int tid = threadIdx.x;

  // --- mask per atom: any neighbor index != -1 ---
  if (tid < AA) {
    const long long* e = edges + ((size_t)s * AA + tid) * DN;
    bool any = false;
#pragma unroll
    for (int d = 0; d < DN; ++d) any |= (e[d] != -1LL);
    maskL[tid] = any ? 1.0f : 0.0f;
  }
  __syncthreads();

  // --- atoms: 128 rows x 31 float2, flat float2 idx = tid + 248*it (coalesced)
  if (tid < 248) {
    const int j = tid % 31;          // fixed feature pair (2j, 2j+1)
    const int g = tid / 31;          // row subgroup 0..7
    const float2* base = (const float2*)atoms + (size_t)s * AA * 31;
    float2 acc = {0.0f, 0.0f};
#pragma unroll 4
    for (int it = 0; it < 16; ++it) {
      const int a = g + 8 * it;
      const float  m = maskL[a];
      const float2 v = base[(size_t)a * 31 + j];
      acc.x += m * v.x;
      acc.y += m * v.y;
    }
    aPart[tid] = acc;
  }
  // --- bonds: 128 rows x 15 float2, flat float2 idx = tid + 240*it (coalesced)
  if (tid < 240) {
    const int j = tid % 15;          // element pair p = 2j, 2j+1 within 30
    const int g = tid / 15;          // row subgroup 0..15
    const float2* base = (const float2*)bonds + (size_t)s * AA * 15;
    float2 acc = {0.0f, 0.0f};
#pragma unroll 4
    for (int it = 0; it < 8; ++it) {
      const int a = g + 16 * it;
      const float  m = maskL[a];
      const float2 v = base[(size_t)a * 15 + j];
      acc.x += m * v.x;
      acc.y += m * v.y;
    }
    bPart[tid] = acc;
  }
  __syncthreads();

  // --- deterministic reductions ---
  if (tid < 31) {                       // atom features
    float2 t = {0.0f, 0.0f};
#pragma unroll
    for (int g = 0; g < 8; ++g) {
      float2 p = aPart[tid + 31 * g];
      t.x += p.x; t.y += p.y;
    }
    accum[2 * tid]     = t.x;
    accum[2 * tid + 1] = t.y;
  }
  if (tid >= 64 && tid < 79) {          // bonds: reduce row-groups per j
    const int j = tid - 64;
    float2 t = {0.0f, 0.0f};
#pragma unroll
    for (int g = 0; g < 16; ++g) {
      float2 p = bPart[j + 15 * g];
      t.x += p.x; t.y += p.y;
    }
    jsum[j] = t;
  }
  __syncthreads();

  if (tid < 3) {                        // bonds: fold j's sharing feature pair
    // j % 3 == tid -> feature pair (2*tid, 2*tid+1)
    float2 t = {0.0f, 0.0f};
#pragma unroll
    for (int i = 0; i < 5; ++i) {
      float2 p = jsum[tid + 3 * i];
      t.x += p.x; t.y += p.y;
    }
    accum[FAF + 2 * tid]     = t.x;
    accum[FAF + 2 * tid + 1] = t.y;
  } else if (tid == 3) {                // masked-atom count (scales bias row)
    float c = 0.0f;
    for (int a = 0; a < AA; ++a) c += maskL[a];
    accum[FAF + FBF] = c;               // index 68
  } else if (tid >= 4 && tid < 7) {
    accum[65 + tid] = 0.0f;             // indices 69..71 = K padding
  }
  __syncthreads();

  if (tid < KP) msum[(size_t)s * KP + tid] = accum[tid];
}

// ---------------------------------------------------------------------------
// Stage 2: out(4096x128) = msum(4096x72) @ wext(72x128) via V_WMMA_F32_16X16X4_F32.
// One wave per 16x16 output tile; K loop = 72/4 = 18 WMMAs.
// A frag (16x4 f32, 2 VGPR): lanes 0-15 -> M=lane, K={k,k+1};
//                            lanes 16-31 -> M=lane-16, K={k+2,k+3}.
// B frag (4x16 f32, 2 VGPR): mirrored K split across lane halves.
// C/D (16x16 f32, 8 VGPR):   VGPR r -> M = r + 8*(lane>=16), N = n0 + (lane&15).
// ---------------------------------------------------------------------------
__global__ void __launch_bounds__(256) gemm_wmma_kernel(
    const float* __restrict__ msum, const float* __restrict__ wext,
    float* __restrict__ out) {
  const int wave = blockIdx.x * (blockDim.x >> 5) + (threadIdx.x >> 5);
  const int lane = threadIdx.x & 31;
  const int nT   = wave & 7;            // 128/16 = 8 tiles in N
  const int mT   = wave >> 3;           // 4096/16 = 256 tiles in M
  const int m0   = mT * 16;
  const int n0   = nT * 16;
  const int half = lane >> 4;           // 0: K+{0,1}, 1: K+{2,3}
  const int lo   = lane & 15;

  const float* aBase = msum + (size_t)(m0 + lo) * KP + 2 * half;
  const float* bBase = wext + (size_t)(2 * half) * FPF + n0 + lo;

  v8f c = {0.0f, 0.0f, 0.0f, 0.0f, 0.0f, 0.0f, 0.0f, 0.0f};
#pragma unroll
  for (int k = 0; k < KP; k += 4) {
    v2f a;
    a.x = aBase[k];
    a.y = aBase[k + 1];
    v2f bf;
    bf.x = bBase[(size_t)k * FPF];
    bf.y = bBase[(size_t)(k + 1) * FPF];
    c = __builtin_amdgcn_wmma_f32_16x16x4_f32(
        /*neg_a=*/false, a, /*neg_b=*/false, bf,
        /*c_mod=*/(short)0, c, /*reuse_a=*/false, /*reuse_b=*/false);
  }

  float* oBase = out + (size_t)(m0 + half * 8) * FPF + n0 + lo;
#pragma unroll
  for (int r = 0; r < 8; ++r) oBase[(size_t)r * FPF] = c[r];
}

// ---------------------------------------------------------------------------
extern "C" void kernel_launch(void* const* d_in, const int* in_sizes, int n_in,
                              void* d_out, int out_size, void* d_ws, size_t ws_size,
                              hipStream_t stream) {
  const float*     atoms = (const float*)d_in[0];       // (S, A, 62) f32
  const float*     bonds = (const float*)d_in[1];       // (S, A, 5, 6) f32
  const long long* edges = (const long long*)d_in[2];   // (S, A, 5) int64
  const float*     W     = (const float*)d_in[3];       // (68, 128) f32
  const float*     b     = (const float*)d_in[4];       // (128,) f32
  float*           out   = (float*)d_out;               // (S, 128) f32

  // Workspace layout: msum (4096 x 72 f32) | wext (72 x 128 f32) ~= 1.19 MB
  float* msum = (float*)d_ws;
  float* wext = msum + (size_t)SS * KP;

  (void)in_sizes; (void)n_in; (void)out_size; (void)ws_size;

  build_wext_kernel<<<(KP * FPF + 255) / 256, 256, 0, stream>>>(W, b, wext);
  reduce_kernel<<<SS, 256, 0, stream>>>(atoms, bonds, edges, msum);
  gemm_wmma_kernel<<<(SS / 16) * (FPF / 16) / 8, 256, 0, stream>>>(msum, wext, out);
}